// Enet_36335423324328
// MI455X (gfx1250) — compile-verified
//
#include <hip/hip_runtime.h>
#include <hip/hip_bf16.h>
#include <math.h>

typedef __attribute__((ext_vector_type(2))) float v2f;
typedef __attribute__((ext_vector_type(8))) float v8f;

__device__ __forceinline__ float leaky(float x) { return x >= 0.0f ? x : 0.1f * x; }

#define KC 64   // K-chunk for the BCN GEMM (multiple of 4)
#define CT 4    // column tiles per workgroup (64 columns)

// ---------------------------------------------------------------------------
// Per-point input MLP: 3 -> 32 -> 32 -> 32, leaky ReLU, output layout (32, N)
// ---------------------------------------------------------------------------
__global__ void k_input_mlp(const float* __restrict__ pc,
                            const float* __restrict__ w1, const float* __restrict__ b1,
                            const float* __restrict__ w2, const float* __restrict__ b2,
                            const float* __restrict__ w3, const float* __restrict__ b3,
                            float* __restrict__ x0, int n) {
  int p = blockIdx.x * blockDim.x + threadIdx.x;
  if (p >= n) return;
  float v0 = pc[p], v1 = pc[n + p], v2 = pc[2 * n + p];
  float l1[32], l2[32];
#pragma unroll
  for (int o = 0; o < 32; ++o)
    l1[o] = leaky(w1[o * 3 + 0] * v0 + w1[o * 3 + 1] * v1 + w1[o * 3 + 2] * v2 + b1[o]);
#pragma unroll 4
  for (int o = 0; o < 32; ++o) {
    float a = b2[o];
#pragma unroll
    for (int c = 0; c < 32; ++c) a += w2[o * 32 + c] * l1[c];
    l2[o] = leaky(a);
  }
#pragma unroll 4
  for (int o = 0; o < 32; ++o) {
    float a = b3[o];
#pragma unroll
    for (int c = 0; c < 32; ++c) a += w3[o * 32 + c] * l2[c];
    x0[(size_t)o * n + p] = leaky(a);
  }
}

// ---------------------------------------------------------------------------
// Splat: lat[(M,C)] += bary * concat(emg, x, 1) per point (segment_sum)
// ---------------------------------------------------------------------------
__global__ void k_splat(const float* __restrict__ x, int Cx,
                        const float* __restrict__ emg, const float* __restrict__ bary,
                        const int* __restrict__ off, float* __restrict__ lat,
                        int n, int C, int M) {
  int p = blockIdx.x * blockDim.x + threadIdx.x;
  if (p >= n) return;
  float w0 = bary[p], w1 = bary[n + p], w2 = bary[2 * n + p], w3 = bary[3 * n + p];
  size_t m0 = (size_t)off[p] * C;
  size_t m1 = (size_t)off[n + p] * C;
  size_t m2 = (size_t)off[2 * n + p] * C;
  size_t m3 = (size_t)off[3 * n + p] * C;
  for (int c = 0; c < C; ++c) {
    float fv;
    if (c < 3)            fv = emg[(size_t)c * n + p];
    else if (c == C - 1)  fv = 1.0f;
    else                  fv = x[(size_t)(c - 3) * n + p];
    atomicAdd(&lat[m0 + c], fv * w0);
    atomicAdd(&lat[m1 + c], fv * w1);
    atomicAdd(&lat[m2 + c], fv * w2);
    atomicAdd(&lat[m3 + c], fv * w3);
  }
}

// ---------------------------------------------------------------------------
// BCN main GEMM: out = leaky(Wb @ leaky(Wa @ gather(lat, nbr) + ba) + bb)
// Workgroup: 64 lattice columns (4 tiles of 16), o0/16 waves, each wave owns a
// 16-row stripe and 4 accumulators (A fragment reused 4x per WMMA K-step).
// Gathered B tile filled with global_load_async_to_lds_b32 (ASYNCcnt path).
// LDS layout (dynamic, starts at wave LDS offset 0 -> async VDST = byte off):
//   [0,       4096)  floats : Btile  KC x 64
//   [4096,    5120)  ints   : nidx   15 x 64 (960 used)
//   [5120, 5120+o0*16) floats: Hlds  o0 x 16 (per column sub-tile)
// ---------------------------------------------------------------------------
__global__ void k_bcn(const float* __restrict__ lat, const int* __restrict__ nbr,
                      const float* __restrict__ Wa, const float* __restrict__ ba,
                      const float* __restrict__ Wb, const float* __restrict__ bb,
                      float* __restrict__ out,
                      int C, int M, int o0) {
  const int F = 15;
  extern __shared__ char smemraw[];
  float* Btile = (float*)smemraw;             // KC*64 floats at LDS offset 0
  int*   nidx  = (int*)(Btile + KC * 64);     // 1024 ints
  float* Hlds  = (float*)(nidx + 1024);       // o0*16 floats

  int vbase = blockIdx.x * (16 * CT);
  int lane  = threadIdx.x & 31;
  int wave  = threadIdx.x >> 5;
  int col   = lane & 15;        // A row index == B/D column index
  int half  = lane >> 4;
  int kk2   = half * 2;
  int K     = C * F;

  for (int t = threadIdx.x; t < 64 * F; t += blockDim.x) {
    int f = t >> 6, v = t & 63;
    nidx[t] = nbr[(size_t)f * M + vbase + v];
  }
  __syncthreads();

  v8f accs[CT];
#pragma unroll
  for (int ct = 0; ct < CT; ++ct)
    accs[ct] = (v8f){0.f, 0.f, 0.f, 0.f, 0.f, 0.f, 0.f, 0.f};

  const float* warow = Wa + (size_t)(wave * 16 + col) * K;

  for (int k0 = 0; k0 < K; k0 += KC) {
    int kc = min(KC, K - k0);
    // Async gather fill of the B tile: memory -> LDS, no VGPR round trip.
    for (int t = threadIdx.x; t < kc * 64; t += blockDim.x) {
      int kl = t >> 6, v = t & 63;
      int k = k0 + kl;
      int c = k / F;
      int f = k - c * F;
      const float* gp = lat + (size_t)nidx[f * 64 + v] * C + c;
      unsigned ldsoff = (unsigned)(t * 4);
      asm volatile("global_load_async_to_lds_b32 %0, %1, off"
                   :: "v"(ldsoff), "v"(gp)
                   : "memory");
    }
    asm volatile("s_wait_asynccnt 0x0" ::: "memory");
    __syncthreads();
#pragma unroll 2
    for (int k = 0; k < kc; k += 4) {
      v2f a;
      a.x = warow[k0 + k + kk2];
      a.y = warow[k0 + k + kk2 + 1];
#pragma unroll
      for (int ct = 0; ct < CT; ++ct) {
        v2f b;
        b.x = Btile[(k + kk2) * 64 + ct * 16 + col];
        b.y = Btile[(k + kk2 + 1) * 64 + ct * 16 + col];
        accs[ct] = __builtin_amdgcn_wmma_f32_16x16x4_f32(false, a, false, b, (short)0,
                                                         accs[ct], false, false);
      }
    }
    __syncthreads();
  }

  // Second GEMM (Wb @ h), one 16-column sub-tile at a time through LDS.
  const float* wbrow = Wb + (size_t)(wave * 16 + col) * o0;
#pragma unroll
  for (int ct = 0; ct < CT; ++ct) {
#pragma unroll
    for (int j = 0; j < 8; ++j) {
      int r = wave * 16 + j + 8 * half;
      Hlds[r * 16 + col] = leaky(accs[ct][j] + ba[r]);
    }
    __syncthreads();
    v8f acc2 = {0.f, 0.f, 0.f, 0.f, 0.f, 0.f, 0.f, 0.f};
#pragma unroll 4
    for (int k = 0; k < o0; k += 4) {
      v2f a, b;
      a.x = wbrow[k + kk2];
      a.y = wbrow[k + kk2 + 1];
      b.x = Hlds[(k + kk2) * 16 + col];
      b.y = Hlds[(k + kk2 + 1) * 16 + col];
      acc2 = __builtin_amdgcn_wmma_f32_16x16x4_f32(false, a, false, b, (short)0, acc2,
                                                   false, false);
    }
#pragma unroll
    for (int j = 0; j < 8; ++j) {
      int r = wave * 16 + j + 8 * half;
      out[(size_t)r * M + vbase + ct * 16 + col] = leaky(acc2[j] + bb[r]);
    }
    __syncthreads();
  }
}

// ---------------------------------------------------------------------------
// Generic WMMA GEMM: out[(o,m)] = A[(o,K)] @ B[(K,m)] + bias  (one wave/tile)
// ---------------------------------------------------------------------------
__global__ void k_gemm(const float* __restrict__ A, const float* __restrict__ B,
                       const float* __restrict__ bias, float* __restrict__ out,
                       int K, int m) {
  int vbase = blockIdx.x * 16;
  int rbase = blockIdx.y * 16;
  int lane  = threadIdx.x & 31;
  int col   = lane & 15;
  int half  = lane >> 4;
  int kk2   = half * 2;
  v8f acc = {0.f, 0.f, 0.f, 0.f, 0.f, 0.f, 0.f, 0.f};
  const float* arow = A + (size_t)(rbase + col) * K;
#pragma unroll 4
  for (int k = 0; k < K; k += 4) {
    v2f a, b;
    a.x = arow[k + kk2];
    a.y = arow[k + kk2 + 1];
    b.x = B[(size_t)(k + kk2) * m + vbase + col];
    b.y = B[(size_t)(k + kk2 + 1) * m + vbase + col];
    acc = __builtin_amdgcn_wmma_f32_16x16x4_f32(false, a, false, b, (short)0, acc,
                                                false, false);
  }
#pragma unroll
  for (int j = 0; j < 8; ++j) {
    int r = rbase + j + 8 * half;
    out[(size_t)r * m + vbase + col] = acc[j] + bias[r];
  }
}

// ---------------------------------------------------------------------------
// BatchNorm stats (per channel over m) and apply (+ReLU, in place)
// ---------------------------------------------------------------------------
__global__ void k_bn_stats(const float* __restrict__ y, float* __restrict__ mu,
                           float* __restrict__ istd, int m) {
  int c = blockIdx.x;
  __shared__ float ss[256], sq[256];
  float s = 0.f, q = 0.f;
  for (int v = threadIdx.x; v < m; v += blockDim.x) {
    float t = y[(size_t)c * m + v];
    s += t; q += t * t;
  }
  ss[threadIdx.x] = s; sq[threadIdx.x] = q;
  __syncthreads();
  for (int st = 128; st > 0; st >>= 1) {
    if (threadIdx.x < st) {
      ss[threadIdx.x] += ss[threadIdx.x + st];
      sq[threadIdx.x] += sq[threadIdx.x + st];
    }
    __syncthreads();
  }
  if (threadIdx.x == 0) {
    float mean = ss[0] / m;
    float var  = sq[0] / m - mean * mean;
    mu[c]   = mean;
    istd[c] = rsqrtf(var + 1e-5f);
  }
}

__global__ void k_bn_apply(float* __restrict__ y, const float* __restrict__ mu,
                           const float* __restrict__ istd, const float* __restrict__ g,
                           const float* __restrict__ be, int m) {
  int i = blockIdx.x * blockDim.x + threadIdx.x;
  int c = i / m;
  float t = y[i];
  y[i] = fmaxf(g[c] * (t - mu[c]) * istd[c] + be[c], 0.f);
}

// ---------------------------------------------------------------------------
// Channel max over m
// ---------------------------------------------------------------------------
__global__ void k_chan_max(const float* __restrict__ y, float* __restrict__ f0, int m) {
  int c = blockIdx.x;
  __shared__ float sm[256];
  float best = -3.402823466e38f;
  for (int v = threadIdx.x; v < m; v += blockDim.x)
    best = fmaxf(best, y[(size_t)c * m + v]);
  sm[threadIdx.x] = best;
  __syncthreads();
  for (int st = 128; st > 0; st >>= 1) {
    if (threadIdx.x < st) sm[threadIdx.x] = fmaxf(sm[threadIdx.x], sm[threadIdx.x + st]);
    __syncthreads();
  }
  if (threadIdx.x == 0) f0[c] = sm[0];
}

// ---------------------------------------------------------------------------
// Tiny FC head + softmax/argmax + Rodrigues rotation -> 23 output floats
// out = [gn_abs(3), gn_sgn(8), e_gn(3), e_T(9)]
// ---------------------------------------------------------------------------
__global__ void k_head(const float* __restrict__ f0,
                       const float* __restrict__ wl1, const float* __restrict__ bl1,
                       const float* __restrict__ wl2, const float* __restrict__ bl2,
                       const float* __restrict__ wl3, const float* __restrict__ bl3,
                       const float* __restrict__ wabs, const float* __restrict__ babs,
                       const float* __restrict__ wsgn, const float* __restrict__ bsgn,
                       float* __restrict__ out) {
  __shared__ float s0[128], s1[128], s2[128], s3[32], sg[8], sa[3];
  int t = threadIdx.x;
  s0[t] = f0[t];
  __syncthreads();
  {
    float a = bl1[t];
    for (int c = 0; c < 128; ++c) a += wl1[t * 128 + c] * s0[c];
    s1[t] = fmaxf(a, 0.f);
  }
  __syncthreads();
  {
    float a = bl2[t];
    for (int c = 0; c < 128; ++c) a += wl2[t * 128 + c] * s1[c];
    s2[t] = fmaxf(a, 0.f);
  }
  __syncthreads();
  if (t < 32) {
    float a = bl3[t];
    for (int c = 0; c < 128; ++c) a += wl3[t * 128 + c] * s2[c];
    s3[t] = fmaxf(a, 0.f);
  }
  __syncthreads();
  if (t < 8) {
    float a = bsgn[t];
    for (int c = 0; c < 32; ++c) a += wsgn[t * 32 + c] * s3[c];
    sg[t] = a;
  }
  if (t < 3) {
    float a = babs[t];
    for (int c = 0; c < 32; ++c) a += wabs[t * 32 + c] * s3[c];
    sa[t] = a;
  }
  __syncthreads();
  if (t == 0) {
    float mx = fmaxf(sa[0], fmaxf(sa[1], sa[2]));
    float e0 = expf(sa[0] - mx), e1 = expf(sa[1] - mx), e2 = expf(sa[2] - mx);
    float inv = 1.f / (e0 + e1 + e2);
    float a0[3] = {e0 * inv, e1 * inv, e2 * inv};
    float nr = 1.f / sqrtf(a0[0] * a0[0] + a0[1] * a0[1] + a0[2] * a0[2]);
    float ga[3] = {a0[0] * nr, a0[1] * nr, a0[2] * nr};
    int idx = 0; float bv = sg[0];
    for (int i = 1; i < 8; ++i) if (sg[i] > bv) { bv = sg[i]; idx = i; }
    float sgn0 = 1.f - 2.f * (float)((idx >> 2) & 1);
    float sgn1 = 1.f - 2.f * (float)((idx >> 1) & 1);
    float sgn2 = 1.f - 2.f * (float)((idx >> 0) & 1);
    float a[3] = {ga[0] * sgn0, ga[1] * sgn1, ga[2] * sgn2};
    float v[3] = {a[1], -a[0], 0.f};   // cross(a, [0,0,1])
    float cc = a[2];                   // dot(a, [0,0,1])
    float vx[9] = {0.f, -v[2], v[1], v[2], 0.f, -v[0], -v[1], v[0], 0.f};
    float vx2[9];
    for (int i = 0; i < 3; ++i)
      for (int j = 0; j < 3; ++j) {
        float s = 0.f;
        for (int k = 0; k < 3; ++k) s += vx[i * 3 + k] * vx[k * 3 + j];
        vx2[i * 3 + j] = s;
      }
    float scale = 1.f / (1.f + cc);
    out[0] = ga[0]; out[1] = ga[1]; out[2] = ga[2];
    for (int i = 0; i < 8; ++i) out[3 + i] = sg[i];
    out[11] = a[0]; out[12] = a[1]; out[13] = a[2];
    for (int i = 0; i < 3; ++i)
      for (int j = 0; j < 3; ++j)
        out[14 + i * 3 + j] = (i == j ? 1.f : 0.f) + vx[i * 3 + j] + vx2[i * 3 + j] * scale;
  }
}

// ---------------------------------------------------------------------------
// Host launcher
// ---------------------------------------------------------------------------
extern "C" void kernel_launch(void* const* d_in, const int* in_sizes, int n_in,
                              void* d_out, int out_size, void* d_ws, size_t ws_size,
                              hipStream_t stream) {
  (void)in_sizes; (void)n_in; (void)out_size; (void)ws_size;

  const float* pc = (const float*)d_in[0];
  const float* emg[5]; const float* bary[5]; const int* off[5]; const int* nbr[5];
  for (int i = 0; i < 5; ++i) {
    emg[i]  = (const float*)d_in[1 + 4 * i];
    bary[i] = (const float*)d_in[2 + 4 * i];
    off[i]  = (const int*)d_in[3 + 4 * i];
    nbr[i]  = (const int*)d_in[4 + 4 * i];
  }
  const float* w_in1 = (const float*)d_in[21]; const float* b_in1 = (const float*)d_in[22];
  const float* w_in2 = (const float*)d_in[23]; const float* b_in2 = (const float*)d_in[24];
  const float* w_in3 = (const float*)d_in[25]; const float* b_in3 = (const float*)d_in[26];
  const float* wa[5]; const float* ba[5]; const float* wb[5]; const float* bb[5];
  for (int i = 0; i < 5; ++i) {
    wa[i] = (const float*)d_in[27 + 4 * i];
    ba[i] = (const float*)d_in[28 + 4 * i];
    wb[i] = (const float*)d_in[29 + 4 * i];
    bb[i] = (const float*)d_in[30 + 4 * i];
  }
  const float* wc1 = (const float*)d_in[47]; const float* bc1 = (const float*)d_in[48];
  const float* wc2 = (const float*)d_in[49]; const float* bc2 = (const float*)d_in[50];
  const float* wc3 = (const float*)d_in[51]; const float* bc3 = (const float*)d_in[52];
  const float* gam1 = (const float*)d_in[53]; const float* bet1 = (const float*)d_in[54];
  const float* gam2 = (const float*)d_in[55]; const float* bet2 = (const float*)d_in[56];
  const float* gam3 = (const float*)d_in[57]; const float* bet3 = (const float*)d_in[58];
  const float* wl1 = (const float*)d_in[59]; const float* bl1 = (const float*)d_in[60];
  const float* wl2 = (const float*)d_in[61]; const float* bl2 = (const float*)d_in[62];
  const float* wl3 = (const float*)d_in[63]; const float* bl3 = (const float*)d_in[64];
  const float* wabs = (const float*)d_in[65]; const float* babs = (const float*)d_in[66];
  const float* wsgn = (const float*)d_in[67]; const float* bsgn = (const float*)d_in[68];

  // Workspace layout (floats)
  float* ws   = (float*)d_ws;
  float* X0   = ws;                    // 32*131072      = 4194304
  float* BUFA = X0 + 4194304;          //                  2097152
  float* BUFB = BUFA + 2097152;        //                  2097152
  float* LAT  = BUFB + 2097152;        // max 36*65536   = 2359296
  float* MU   = LAT + 2359296;         // 128
  float* ISTD = MU + 128;              // 128
  float* F0   = ISTD + 128;            // 128

  const int N = 131072;
  const int MSa[5]  = {65536, 32768, 16384, 8192, 4096};
  const int NSa[5]  = {131072, 65536, 32768, 16384, 8192};
  const int CINa[5] = {36, 36, 68, 132, 260};
  const int O0a[5]  = {32, 64, 128, 256, 256};

  k_input_mlp<<<N / 256, 256, 0, stream>>>(pc, w_in1, b_in1, w_in2, b_in2, w_in3, b_in3,
                                           X0, N);

  const float* xin = X0;
  float* bufs[2] = {BUFA, BUFB};
  for (int s = 0; s < 5; ++s) {
    int n = NSa[s], M = MSa[s], C = CINa[s], o0 = O0a[s];
    hipMemsetAsync(LAT, 0, (size_t)C * M * sizeof(float), stream);
    k_splat<<<(n + 255) / 256, 256, 0, stream>>>(xin, C - 4, emg[s], bary[s], off[s],
                                                 LAT, n, C, M);
    float* outp = bufs[s & 1];
    int nwaves = o0 / 16;
    size_t smem = (size_t)(KC * 64) * 4 + 1024 * 4 + (size_t)o0 * 16 * 4;
    k_bcn<<<M / (16 * CT), nwaves * 32, smem, stream>>>(LAT, nbr[s], wa[s], ba[s],
                                                        wb[s], bb[s], outp, C, M, o0);
    xin = outp;
  }
  // stage 5 output: BUFA, shape (256, 4096)

  const int m = 4096;
  dim3 gconv(m / 16, 128 / 16);
  // conv1: 256 -> 128
  k_gemm<<<gconv, 32, 0, stream>>>(wc1, BUFA, bc1, BUFB, 256, m);
  k_bn_stats<<<128, 256, 0, stream>>>(BUFB, MU, ISTD, m);
  k_bn_apply<<<(128 * m) / 256, 256, 0, stream>>>(BUFB, MU, ISTD, gam1, bet1, m);
  // conv2: 128 -> 128
  k_gemm<<<gconv, 32, 0, stream>>>(wc2, BUFB, bc2, BUFA, 128, m);
  k_bn_stats<<<128, 256, 0, stream>>>(BUFA, MU, ISTD, m);
  k_bn_apply<<<(128 * m) / 256, 256, 0, stream>>>(BUFA, MU, ISTD, gam2, bet2, m);
  // conv3: 128 -> 128
  k_gemm<<<gconv, 32, 0, stream>>>(wc3, BUFA, bc3, BUFB, 128, m);
  k_bn_stats<<<128, 256, 0, stream>>>(BUFB, MU, ISTD, m);
  k_bn_apply<<<(128 * m) / 256, 256, 0, stream>>>(BUFB, MU, ISTD, gam3, bet3, m);

  k_chan_max<<<128, 256, 0, stream>>>(BUFB, F0, m);
  k_head<<<1, 128, 0, stream>>>(F0, wl1, bl1, wl2, bl2, wl3, bl3, wabs, babs, wsgn, bsgn,
                                (float*)d_out);
}